// TaggerHead_30442728194182
// MI455X (gfx1250) — compile-verified
//
#include <hip/hip_runtime.h>
#include <hip/hip_bf16.h>

// Problem constants (from reference): B=64, L=512, H=1024, T=128
#define BB 64
#define LL 512
#define HH 1024
#define TT 128

typedef __attribute__((ext_vector_type(2))) float v2f;
typedef __attribute__((ext_vector_type(8))) float v8f;

// ---------------------------------------------------------------------------
// Kernel 1: logits[m, n] = sum_k E[m,k] * W[n,k] + bias[n]
//   M = B*L = 32768, N = T = 128, K = H = 1024
//   FP32 WMMA (v_wmma_f32_16x16x4_f32) to preserve exact fp32 argmax behavior.
//   Grid: M/16 blocks; 256 threads = 8 waves; wave w owns N-tile [16w, 16w+16).
//
// A-matrix (16x4 f32) per ISA layout: lane l holds row m = l&15,
//   v0/v1 = K = k0 + 2*(l>>4) + {0,1}   -> one b64 load
// B-matrix (4x16 f32): lane l holds col n = l&15,
//   v0/v1 = K = k0 + 2*(l>>4) + {0,1}   -> one b64 load from W[n*H + ...]
// C/D (16x16 f32): VGPR r = row m_base + r + 8*(l>>4), col n_base + (l&15)
// ---------------------------------------------------------------------------
__global__ void __launch_bounds__(256)
tagger_gemm_wmma(const float* __restrict__ E, const float* __restrict__ W,
                 const float* __restrict__ bias, float* __restrict__ logits) {
    const int lane = threadIdx.x & 31;
    const int wave = threadIdx.x >> 5;
    const int m_base = blockIdx.x * 16;
    const int n_base = wave * 16;

    const int half = lane >> 4;          // 0: lanes 0-15, 1: lanes 16-31
    const int mrow = m_base + (lane & 15);
    const int ncol = n_base + (lane & 15);
    const int koff = half * 2;

    const float* arow = E + (size_t)mrow * HH + koff;
    const float* brow = W + (size_t)ncol * HH + koff;

    v8f c = {};
#pragma unroll 8
    for (int k0 = 0; k0 < HH; k0 += 4) {
        v2f a = *(const v2f*)(arow + k0);
        v2f b = *(const v2f*)(brow + k0);
        // 8-arg form: (neg_a, A, neg_b, B, c_mod, C, reuse_a, reuse_b)
        c = __builtin_amdgcn_wmma_f32_16x16x4_f32(false, a, false, b,
                                                  (short)0, c, false, false);
    }

    const float bn = bias[ncol];
    const int rbase = m_base + half * 8;
#pragma unroll
    for (int r = 0; r < 8; ++r) {
        logits[(size_t)(rbase + r) * TT + ncol] = c[r] + bn;
    }
}

// ---------------------------------------------------------------------------
// Kernel 2: Viterbi forward scan + backtrace, one workgroup per batch element.
//   LDS: transitions (64 KB) + score (512 B) + partials (4 KB) + hist (64 KB)
//   512 threads: thread (j = tid&127, g = tid>>7) reduces i in [32g, 32g+32)
//   with strict '>' in ascending i -> first-max semantics (matches jnp.argmax).
// ---------------------------------------------------------------------------
#define SMEM_TRANS_BYTES   (TT * TT * 4)        // 65536
#define SMEM_SCORE_BYTES   (TT * 4)             // 512
#define SMEM_PBEST_BYTES   (512 * 4)            // 2048
#define SMEM_PARG_BYTES    (512 * 4)            // 2048
#define SMEM_HIST_BYTES    ((LL - 1) * TT)      // 65408
#define SMEM_TOTAL_BYTES   (SMEM_TRANS_BYTES + SMEM_SCORE_BYTES + \
                            SMEM_PBEST_BYTES + SMEM_PARG_BYTES + SMEM_HIST_BYTES)

__global__ void __launch_bounds__(512)
tagger_viterbi(const float* __restrict__ logits,
               const unsigned char* __restrict__ mask,
               const float* __restrict__ startT,
               const float* __restrict__ endT,
               const float* __restrict__ trans,
               int* __restrict__ paths) {
    extern __shared__ char smem[];
    float*         sTrans = (float*)smem;
    float*         sScore = (float*)(smem + SMEM_TRANS_BYTES);
    float*         sBest  = (float*)(smem + SMEM_TRANS_BYTES + SMEM_SCORE_BYTES);
    int*           sArg   = (int*)  (smem + SMEM_TRANS_BYTES + SMEM_SCORE_BYTES + SMEM_PBEST_BYTES);
    unsigned char* sHist  = (unsigned char*)(smem + SMEM_TRANS_BYTES + SMEM_SCORE_BYTES +
                                             SMEM_PBEST_BYTES + SMEM_PARG_BYTES);

    const int b   = blockIdx.x;
    const int tid = threadIdx.x;
    const int j   = tid & (TT - 1);
    const int g   = tid >> 7;           // 0..3

    // Stage transitions[i][j] into LDS (row-major: thread j reads column j,
    // consecutive lanes -> consecutive banks).
    for (int idx = tid; idx < TT * TT; idx += 512) sTrans[idx] = trans[idx];

    const float* elog = logits + (size_t)b * LL * TT;
    if (g == 0) sScore[j] = startT[j] + elog[j];
    __syncthreads();

    for (int t = 1; t < LL; ++t) {
        float em = 0.0f;
        if (g == 0) em = elog[(size_t)t * TT + j];       // overlaps with partials
        const unsigned int m = mask[(size_t)b * LL + t]; // uniform per block

        // Partial max-plus reduction over i in [32g, 32g+32)
        float best = -__builtin_inff();
        int   arg  = 0;
        const int ibase = g * 32;
#pragma unroll 8
        for (int ii = 0; ii < 32; ++ii) {
            const int i = ibase + ii;
            const float v = sScore[i] + sTrans[i * TT + j];
            if (v > best) { best = v; arg = i; }
        }
        sBest[tid] = best;
        sArg[tid]  = arg;
        __syncthreads();

        if (g == 0) {
            float bb = sBest[j];
            int   aa = sArg[j];
#pragma unroll
            for (int gg = 1; gg < 4; ++gg) {
                const float v = sBest[j + gg * TT];
                if (v > bb) { bb = v; aa = sArg[j + gg * TT]; }
            }
            sHist[(t - 1) * TT + j] = (unsigned char)aa;   // recorded regardless of mask
            const float old = sScore[j];
            sScore[j] = m ? (bb + em) : old;
        }
        __syncthreads();
    }

    // Final scores + argmax over tags, then backtrace (all LDS-resident).
    if (g == 0) {
        sBest[j] = sScore[j] + endT[j];
    }
    __syncthreads();

    if (tid == 0) {
        float bb  = sBest[0];
        int   cur = 0;
        for (int i = 1; i < TT; ++i) {
            if (sBest[i] > bb) { bb = sBest[i]; cur = i; }
        }
        int* prow = paths + (size_t)b * LL;
        prow[LL - 1] = cur;
        for (int tt = LL - 2; tt >= 0; --tt) {
            const int prev = sHist[tt * TT + cur];
            if (mask[(size_t)b * LL + tt + 1]) cur = prev;
            prow[tt] = cur;
        }
    }
}

// ---------------------------------------------------------------------------
// Launcher. Inputs (setup_inputs order):
//   0: emissions f32 [B,L,H]   1: mask bool(u8) [B,L]   2: W f32 [T,H]
//   3: b f32 [T]  4: start_transitions f32 [T]  5: end_transitions f32 [T]
//   6: transitions f32 [T,T]
// Output: int32 paths [B,L]. Workspace: logits f32 [B*L, T] = 16 MB.
// ---------------------------------------------------------------------------
extern "C" void kernel_launch(void* const* d_in, const int* in_sizes, int n_in,
                              void* d_out, int out_size, void* d_ws, size_t ws_size,
                              hipStream_t stream) {
    const float*         E      = (const float*)d_in[0];
    const unsigned char* mask   = (const unsigned char*)d_in[1];
    const float*         W      = (const float*)d_in[2];
    const float*         bias   = (const float*)d_in[3];
    const float*         startT = (const float*)d_in[4];
    const float*         endT   = (const float*)d_in[5];
    const float*         trans  = (const float*)d_in[6];
    int*   paths  = (int*)d_out;
    float* logits = (float*)d_ws;   // B*L*T floats = 16 MB

    // Allow ~132 KB dynamic LDS (CDNA5 WGP has 320 KB).
    (void)hipFuncSetAttribute((const void*)tagger_viterbi,
                              hipFuncAttributeMaxDynamicSharedMemorySize,
                              SMEM_TOTAL_BYTES);

    // Phase 1: WMMA GEMM -> logits (bandwidth-bound, ~145 MB @ 23.3 TB/s)
    dim3 g1((BB * LL) / 16);
    tagger_gemm_wmma<<<g1, 256, 0, stream>>>(E, W, bias, logits);

    // Phase 2: Viterbi scan + backtrace, one WGP-resident block per batch elem
    tagger_viterbi<<<dim3(BB), 512, SMEM_TOTAL_BYTES, stream>>>(
        logits, mask, startT, endT, trans, paths);
}